// CosineHammingLSH_90675349553507
// MI455X (gfx1250) — compile-verified
//
#include <hip/hip_runtime.h>

// LSH bucketed attention: per bucket O = exp(Q K^T) W, all tiles 64x64x64, f32.
// Memory-bound (272 MB @ 23.3 TB/s ~ 12 us floor), so f32 end-to-end with
// V_WMMA_F32_16X16X4_F32 for both GEMMs; one workgroup (8 wave32) per bucket.

typedef __attribute__((ext_vector_type(2))) float v2f;
typedef __attribute__((ext_vector_type(8))) float v8f;

#define BUCKET 64
#define DIM    64
#define LSTR   68   // padded LDS row stride (floats): bank stride 4 -> conflict-free WMMA fetch

__global__ __launch_bounds__(256)
void lsh_bucket_attn_kernel(const float* __restrict__ q,
                            const float* __restrict__ k,
                            const float* __restrict__ w,
                            const int*   __restrict__ kidx,
                            const int*   __restrict__ qidx,
                            float*       __restrict__ out,
                            int N)
{
    __shared__ float Qs[BUCKET * LSTR];  // gathered Q rows; reused for O before scatter
    __shared__ float Ks[BUCKET * LSTR];  // gathered K rows
    __shared__ float Wt[DIM   * LSTR];   // gathered W, transposed: Wt[v][kk]
    __shared__ float Ss[BUCKET * LSTR];  // exp(Q K^T)
    __shared__ int   qi[BUCKET];         // cached Q permutation for scatter

    const int t        = threadIdx.x;
    const int nblocks  = N / BUCKET;
    const int bh       = blockIdx.x / nblocks;
    const int blk      = blockIdx.x % nblocks;
    const long base    = (long)bh * N + (long)blk * BUCKET;

    // ---- gather permuted rows into LDS (each thread: one row-quarter = 16 floats) ----
    {
        const int r  = t >> 2;          // bucket row 0..63
        const int cq = (t & 3) * 16;    // column quarter base
        const int qrow = qidx[base + r];
        const int krow = kidx[base + r];
        if (t < BUCKET) qi[t] = qidx[base + t];

        const float4* qsrc = (const float4*)(q + ((long)bh * N + qrow) * DIM + cq);
        const float4* ksrc = (const float4*)(k + ((long)bh * N + krow) * DIM + cq);
        const float4* wsrc = (const float4*)(w + ((long)bh * N + krow) * DIM + cq);
        #pragma unroll
        for (int j = 0; j < 4; ++j) {
            float4 qv = qsrc[j];
            float4 kv = ksrc[j];
            float4 wv = wsrc[j];
            *(float4*)(Qs + r * LSTR + cq + 4 * j) = qv;
            *(float4*)(Ks + r * LSTR + cq + 4 * j) = kv;
            // transpose W so second-GEMM B-operand is a contiguous float2 per lane
            Wt[(cq + 4 * j + 0) * LSTR + r] = wv.x;
            Wt[(cq + 4 * j + 1) * LSTR + r] = wv.y;
            Wt[(cq + 4 * j + 2) * LSTR + r] = wv.z;
            Wt[(cq + 4 * j + 3) * LSTR + r] = wv.w;
        }
    }
    __syncthreads();

    const int lane = t & 31;
    const int wave = t >> 5;        // 0..7
    const int l15  = lane & 15;     // M (A) / N (B,C,D) within tile
    const int koff = (lane >> 4) * 2; // K sub-offset for A/B operands
    const int mrow = (lane >> 4) * 8; // C/D row offset for upper half-wave

    // ---- stage 1: S = exp(Q K^T), two 16x16 tiles per wave ----
    #pragma unroll
    for (int i = 0; i < 2; ++i) {
        const int tile = wave * 2 + i;          // 0..15
        const int tm = (tile >> 2) * 16;
        const int tn = (tile & 3) * 16;
        v8f acc = {};
        #pragma unroll
        for (int ks = 0; ks < 16; ++ks) {
            const int kk = ks * 4 + koff;
            v2f a = *(const v2f*)(Qs + (tm + l15) * LSTR + kk);  // A[m][kk..kk+1]
            v2f b = *(const v2f*)(Ks + (tn + l15) * LSTR + kk);  // B[kk..kk+1][n] = K[n][kk..]
            acc = __builtin_amdgcn_wmma_f32_16x16x4_f32(
                false, a, false, b, (short)0, acc, false, false);
        }
        #pragma unroll
        for (int r = 0; r < 8; ++r)
            Ss[(tm + r + mrow) * LSTR + tn + l15] = __expf(acc[r]);
    }
    __syncthreads();

    // ---- stage 2: O = S W, write into Qs (Q consumed) ----
    #pragma unroll
    for (int i = 0; i < 2; ++i) {
        const int tile = wave * 2 + i;
        const int tm = (tile >> 2) * 16;
        const int tn = (tile & 3) * 16;
        v8f acc = {};
        #pragma unroll
        for (int ks = 0; ks < 16; ++ks) {
            const int kk = ks * 4 + koff;
            v2f a = *(const v2f*)(Ss + (tm + l15) * LSTR + kk);  // A[m][kk..kk+1]
            v2f b = *(const v2f*)(Wt + (tn + l15) * LSTR + kk);  // B[kk..kk+1][v] = Wt[v][kk..]
            acc = __builtin_amdgcn_wmma_f32_16x16x4_f32(
                false, a, false, b, (short)0, acc, false, false);
        }
        #pragma unroll
        for (int r = 0; r < 8; ++r)
            Qs[(tm + r + mrow) * LSTR + tn + l15] = acc[r];
    }
    __syncthreads();

    // ---- scatter: out[b,h, Q_sort_idx[base+r], :] = O[r][:] (vectorized) ----
    {
        const int r  = t >> 2;
        const int cq = (t & 3) * 16;
        const int orow = qi[r];
        float* dst = out + ((long)bh * N + orow) * DIM + cq;
        #pragma unroll
        for (int j = 0; j < 4; ++j)
            *(float4*)(dst + 4 * j) = *(const float4*)(Qs + r * LSTR + cq + 4 * j);
    }
}

extern "C" void kernel_launch(void* const* d_in, const int* in_sizes, int n_in,
                              void* d_out, int out_size, void* d_ws, size_t ws_size,
                              hipStream_t stream) {
    const float* q    = (const float*)d_in[0];
    const float* k    = (const float*)d_in[1];
    const float* w    = (const float*)d_in[2];
    const int*   kidx = (const int*)d_in[3];
    const int*   qidx = (const int*)d_in[4];
    float*       out  = (float*)d_out;

    const int N  = 4096;                 // reference: B=4, H=16, N=4096, D=64
    const int BH = in_sizes[3] / N;      // B*H from K_sort_idx element count
    const int nblocks = N / BUCKET;      // 64 buckets per head

    dim3 grid(BH * nblocks);             // 4096 workgroups, one per bucket
    dim3 block(256);                     // 8 wave32 per workgroup
    lsh_bucket_attn_kernel<<<grid, block, 0, stream>>>(q, k, w, kidx, qidx, out, N);
}